// SymmetricCosineSimilarityLoss_6511170421410
// MI455X (gfx1250) — compile-verified
//
#include <hip/hip_runtime.h>

// ---------------------------------------------------------------------------
// Symmetric cosine-similarity loss, MI455X (gfx1250, wave32, WMMA).
//   pred, target: (B=16, C=128, H=48, W=48) fp32  ->  x,y: (B, N=2304, C=128)
//   loss = mean_{b,n,m} (cos[b,n,m] - cos[b,m,n])^2
// Strategy: triangular 16x16 tile pairs (I<=J); per wave two fp32 WMMA GEMM
// chains D1 = X_I Y_J^T, D2 = X_J Y_I^T over K=C=128 (32 x v_wmma_f32_16x16x4_f32
// each); norms accumulated from the same fragments; c2 transposed via LDS.
// ---------------------------------------------------------------------------

typedef __attribute__((ext_vector_type(2))) float v2f;
typedef __attribute__((ext_vector_type(8))) float v8f;

#define B_      16
#define C_      128
#define HW_     2304          // 48*48
#define T_      144           // HW_/16 tiles per dimension
#define NPAIRS_ 10440         // T*(T+1)/2
#define COS_EPS 1e-6f

__global__ void zero_out_kernel(float* out) { out[0] = 0.0f; }

__global__ __launch_bounds__(32)
void sym_cos_loss_kernel(const float* __restrict__ pred,
                         const float* __restrict__ targ,
                         float* __restrict__ out)
{
    __shared__ float ldsT[16 * 16];   // one wave per workgroup

    const int lane = threadIdx.x;     // 0..31
    const int col  = lane & 15;       // column index within tile (m for A, n/j for B)
    const int half = lane >> 4;       // which K-pair this lane carries

    // Decode triangular pair index -> (I, J), I <= J (uniform across wave).
    int p = blockIdx.x;
    int I = 0;
    while (p >= (T_ - I)) { p -= (T_ - I); ++I; }
    const int J = I + p;
    const int b = blockIdx.y;

    const float* __restrict__ x = pred + (size_t)b * C_ * HW_;  // x[c*HW + n]
    const float* __restrict__ y = targ + (size_t)b * C_ * HW_;

    const int nI = I * 16 + col;
    const int nJ = J * 16 + col;

    v8f acc1 = {};                 // D1 = X_I * Y_J^T   (16x16 f32)
    v8f acc2 = {};                 // D2 = X_J * Y_I^T
    float sxI = 0.f, syJ = 0.f, sxJ = 0.f, syI = 0.f;   // per-column sum of squares

    #pragma unroll 4
    for (int k0 = 0; k0 < C_; k0 += 4) {
        const int r0 = k0 + 2 * half;                  // this lane's first K row
        // A fragment (16x4): lane = m + 16*(k/2), vgpr = k&1  -> rows r0, r0+1 at col m
        // B fragment (4x16): lane = n + 16*(k/2), vgpr = k&1  -> rows r0, r0+1 at col n
        v2f a1, b1, a2, b2;
        a1.x = x[(size_t)r0 * HW_ + nI];  a1.y = x[(size_t)(r0 + 1) * HW_ + nI];
        b1.x = y[(size_t)r0 * HW_ + nJ];  b1.y = y[(size_t)(r0 + 1) * HW_ + nJ];
        a2.x = x[(size_t)r0 * HW_ + nJ];  a2.y = x[(size_t)(r0 + 1) * HW_ + nJ];
        b2.x = y[(size_t)r0 * HW_ + nI];  b2.y = y[(size_t)(r0 + 1) * HW_ + nI];

        sxI += a1.x * a1.x + a1.y * a1.y;   // norms for free from fragment data
        syJ += b1.x * b1.x + b1.y * b1.y;
        sxJ += a2.x * a2.x + a2.y * a2.y;
        syI += b2.x * b2.x + b2.y * b2.y;

        // (neg_a, A, neg_b, B, c_mod, C, reuse_a, reuse_b)
        acc1 = __builtin_amdgcn_wmma_f32_16x16x4_f32(false, a1, false, b1,
                                                     (short)0, acc1, false, false);
        acc2 = __builtin_amdgcn_wmma_f32_16x16x4_f32(false, a2, false, b2,
                                                     (short)0, acc2, false, false);
    }

    // Fold the two K-halves: lane L and L^16 accumulated disjoint K subsets of
    // the same column (L&15). Afterwards lane L holds the full sum for col L&15.
    sxI += __shfl_xor(sxI, 16, 32);
    syJ += __shfl_xor(syJ, 16, 32);
    sxJ += __shfl_xor(sxJ, 16, 32);
    syI += __shfl_xor(syI, 16, 32);

    const float nxI_own = sqrtf(sxI);   // ||x_{I*16 + (lane&15)}||
    const float nyJ_own = sqrtf(syJ);   // ||y_{J*16 + (lane&15)}||
    const float nxJ_own = sqrtf(sxJ);
    const float nyI_own = sqrtf(syI);

    // C/D layout: lane L holds D(m, j) for j = L&15, m = 8*(L>>4) + r, r=0..7.
    const int mbase = 8 * half;
    float c1[8], c2v[8];
    #pragma unroll
    for (int r = 0; r < 8; ++r) {
        const float nxI_m = __shfl(nxI_own, mbase + r, 32);   // row norm broadcast
        const float nxJ_m = __shfl(nxJ_own, mbase + r, 32);
        const float d1 = fmaxf(nxI_m * nyJ_own, COS_EPS);
        const float d2 = fmaxf(nxJ_m * nyI_own, COS_EPS);
        c1[r]  = acc1[r] / d1;    // cos( x_{I,m}, y_{J,j} )
        c2v[r] = acc2[r] / d2;    // cos( x_{J,m}, y_{I,j} )
    }

    // Transpose c2 through LDS (intra-wave, single 16x16 tile).
    #pragma unroll
    for (int r = 0; r < 8; ++r)
        ldsT[(mbase + r) * 16 + col] = c2v[r];
    __syncthreads();   // single-wave workgroup: orders LDS write -> read

    float term = 0.f;
    #pragma unroll
    for (int r = 0; r < 8; ++r) {
        const float c2t = ldsT[col * 16 + (mbase + r)];   // c2[j, i]
        const float d = c1[r] - c2t;                      // c[n,m] - c[m,n]
        term += d * d;
    }

    // Wave reduction of the 16x16 tile sum.
    #pragma unroll
    for (int off = 16; off > 0; off >>= 1)
        term += __shfl_xor(term, off, 32);

    if (lane == 0) {
        const float w = (I == J) ? 1.0f : 2.0f;   // (I,J) and (J,I) tiles are equal
        const float scale = 1.0f / ((float)B_ * (float)HW_ * (float)HW_);
        atomicAdd(out, term * w * scale);
    }
}

extern "C" void kernel_launch(void* const* d_in, const int* in_sizes, int n_in,
                              void* d_out, int out_size, void* d_ws, size_t ws_size,
                              hipStream_t stream) {
    (void)in_sizes; (void)n_in; (void)out_size; (void)d_ws; (void)ws_size;
    const float* pred = (const float*)d_in[0];
    const float* targ = (const float*)d_in[1];
    float* out = (float*)d_out;

    zero_out_kernel<<<1, 1, 0, stream>>>(out);

    dim3 grid(NPAIRS_, B_);
    dim3 block(32);
    sym_cos_loss_kernel<<<grid, block, 0, stream>>>(pred, targ, out);
}